// pForward_76106820485730
// MI455X (gfx1250) — compile-verified
//
#include <hip/hip_runtime.h>
#include <hip/hip_bf16.h>

// ---------------------------------------------------------------------------
// Problem constants (from the reference): B=8, D=16, N=384, M=384, T=384
// Pipeline:
//   K1: Ax[b,d,t,m] = sum_n fmt_sub[b,d,t,n] * i_coil[b,d,n,m]   (complex f32 GEMM,
//       fmt_sub and i_coil computed on the fly -> LDS, WMMA f32 16x16x4 core)
//   K2: out[b,m,d,t] = fftshift(DFT_16(fftshift(Ax[b,:,t,m])))   (direct 16-pt DFT)
// ---------------------------------------------------------------------------

#define Bb 8
#define Dd 16
#define Nn 384
#define Mm 384
#define Tt 384

#define KB   32   // k-chunk staged in LDS
#define TILE 64   // 64x64 (T x M) output tile per workgroup

typedef float v2f __attribute__((ext_vector_type(2)));
typedef float v8f __attribute__((ext_vector_type(8)));

#define TWO_PI 6.283185307179586f

// ---------------------------------------------------------------------------
// Kernel 1: fused encode + complex GEMM using V_WMMA_F32_16X16X4_F32
// grid: (M/64, T/64, B*D), block: 256 threads (8 waves of 32)
// ---------------------------------------------------------------------------
__global__ __launch_bounds__(256) void fmt_gemm_wmma_kernel(
    const float2* __restrict__ o1,    // [B,D,N,M] complex
    const float2* __restrict__ c1,    // [B,M,D,N] complex
    const float*  __restrict__ m1,    // [B,D,T]
    const float*  __restrict__ f1,    // [B,D,N] (last dim 1 squeezed)
    const float2* __restrict__ fmt,   // [T,N] complex
    const float*  __restrict__ tl,    // [T]
    const int*    __restrict__ updown,// scalar
    float2*       __restrict__ Ax)    // [B,D,T,M] complex (workspace)
{
    __shared__ float Ar[TILE][KB + 1];
    __shared__ float Ai[TILE][KB + 1];
    __shared__ float Br[KB][TILE + 1];
    __shared__ float Bi[KB][TILE + 1];

    const int bd  = blockIdx.z;            // b*D + d
    const int b   = bd >> 4;
    const int d   = bd & 15;
    const int t0  = blockIdx.y * TILE;
    const int m0  = blockIdx.x * TILE;
    const int tid = threadIdx.x;
    const int lane = tid & 31;
    const int wave = tid >> 5;
    const bool up  = (updown[0] != 0);

    // wave -> two 16x16 subtiles: rows t0 + 16*(wave>>1), cols m0 + 32*(wave&1) + {0,16}
    const int wr = (wave >> 1) << 4;       // 0,16,32,48
    const int wc = (wave & 1) << 5;        // 0,32

    v8f accRe0 = {}, accIm0 = {}, accRe1 = {}, accIm1 = {};

    const float* f1bd  = f1 + (b * Dd + d) * Nn;
    const float* m1bd  = m1 + (b * Dd + d) * Tt;
    const float2* o1bd = o1 + (size_t)(b * Dd + d) * Nn * Mm;

    for (int k0 = 0; k0 < Nn; k0 += KB) {
        // ---- stage A-panel: fmt_sub rows t0..t0+63, cols k0..k0+31 (2048 elems, 8/thread)
        #pragma unroll
        for (int i = 0; i < 8; ++i) {
            const int e  = tid + (i << 8);
            const int kk = e & (KB - 1);
            const int ttl = e >> 5;            // 0..63
            const int t  = t0 + ttl;
            const int n  = k0 + kk;
            const float te = up ? tl[t] : tl[Tt - 1 - t];
            const float ph = TWO_PI * te * f1bd[n];
            float s, c;
            __sincosf(ph, &s, &c);
            const float2 fv = fmt[t * Nn + n];
            const float  mv = m1bd[t];
            Ar[ttl][kk] = (fv.x * c - fv.y * s) * mv;   // Re(fmt * e^{i ph}) * mask
            Ai[ttl][kk] = (fv.x * s + fv.y * c) * mv;   // Im(fmt * e^{i ph}) * mask
        }
        // ---- stage B-panel: i_coil rows k0..k0+31, cols m0..m0+63 (2048 elems, 8/thread)
        #pragma unroll
        for (int i = 0; i < 8; ++i) {
            const int e    = tid + (i << 8);
            const int mcol = e & (TILE - 1);
            const int nn   = e >> 6;           // 0..31
            const int n    = k0 + nn;
            const int m    = m0 + mcol;
            const float2 ov = o1bd[(size_t)n * Mm + m];
            const float2 cv = c1[((size_t)(b * Mm + m) * Dd + d) * Nn + n];
            Br[nn][mcol] = ov.x * cv.x - ov.y * cv.y;
            Bi[nn][mcol] = ov.x * cv.y + ov.y * cv.x;
        }
        // hint next o1 panel into cache (global_prefetch_b8)
        if (k0 + KB < Nn) {
            const int mcol = tid & (TILE - 1);
            const int nn   = tid >> 6;
            __builtin_prefetch(&o1bd[(size_t)(k0 + KB + nn) * Mm + (m0 + mcol)], 0, 1);
        }
        __syncthreads();

        // ---- WMMA core: complex MAC over the 32-wide k chunk, 4 k per step
        #pragma unroll
        for (int kk = 0; kk < KB; kk += 4) {
            // A fragment (16x4 f32): lanes 0-15 carry K=kk,kk+1; lanes 16-31 K=kk+2,kk+3
            const int arow = wr + (lane & 15);
            const int kb   = kk + ((lane >> 4) << 1);
            v2f ar, ai;
            ar[0] = Ar[arow][kb];     ar[1] = Ar[arow][kb + 1];
            ai[0] = Ai[arow][kb];     ai[1] = Ai[arow][kb + 1];
            const v2f nai = -ai;      // f32 WMMA NEG only covers C -> negate Ai in VALU
            // B fragments (4x16 f32), two subtiles
            const int bc = wc + (lane & 15);
            v2f br0, bi0, br1, bi1;
            br0[0] = Br[kb][bc];          br0[1] = Br[kb + 1][bc];
            bi0[0] = Bi[kb][bc];          bi0[1] = Bi[kb + 1][bc];
            br1[0] = Br[kb][bc + 16];     br1[1] = Br[kb + 1][bc + 16];
            bi1[0] = Bi[kb][bc + 16];     bi1[1] = Bi[kb + 1][bc + 16];

            accRe0 = __builtin_amdgcn_wmma_f32_16x16x4_f32(false, ar,  false, br0, (short)0, accRe0, false, false);
            accRe0 = __builtin_amdgcn_wmma_f32_16x16x4_f32(false, nai, false, bi0, (short)0, accRe0, false, false);
            accIm0 = __builtin_amdgcn_wmma_f32_16x16x4_f32(false, ar,  false, bi0, (short)0, accIm0, false, false);
            accIm0 = __builtin_amdgcn_wmma_f32_16x16x4_f32(false, ai,  false, br0, (short)0, accIm0, false, false);

            accRe1 = __builtin_amdgcn_wmma_f32_16x16x4_f32(false, ar,  false, br1, (short)0, accRe1, false, false);
            accRe1 = __builtin_amdgcn_wmma_f32_16x16x4_f32(false, nai, false, bi1, (short)0, accRe1, false, false);
            accIm1 = __builtin_amdgcn_wmma_f32_16x16x4_f32(false, ar,  false, bi1, (short)0, accIm1, false, false);
            accIm1 = __builtin_amdgcn_wmma_f32_16x16x4_f32(false, ai,  false, br1, (short)0, accIm1, false, false);
        }
        __syncthreads();
    }

    // ---- epilogue: C/D layout: VGPR r -> rows r (lanes 0-15) and r+8 (lanes 16-31)
    const int crow = t0 + wr + ((lane >> 4) << 3);
    const int ccol = m0 + wc + (lane & 15);
    float2* axbd = Ax + (size_t)bd * Tt * Mm;
    #pragma unroll
    for (int r = 0; r < 8; ++r) {
        float2 v0; v0.x = accRe0[r]; v0.y = accIm0[r];
        axbd[(size_t)(crow + r) * Mm + ccol] = v0;
        float2 v1; v1.x = accRe1[r]; v1.y = accIm1[r];
        axbd[(size_t)(crow + r) * Mm + ccol + 16] = v1;
    }
}

// ---------------------------------------------------------------------------
// Kernel 2: fftshift -> 16-pt DFT along D -> fftshift, with [B,D,T,M]->[B,M,D,T]
// one thread per (b,t,m); both shifts folded into index arithmetic
// ---------------------------------------------------------------------------
__global__ __launch_bounds__(256) void dfft16_shift_kernel(
    const float2* __restrict__ Ax,   // [B,D,T,M]
    float2*       __restrict__ out)  // [B,M,D,T]
{
    __shared__ float2 tw[16];
    if (threadIdx.x < 16) {
        float s, c;
        __sincosf(-TWO_PI * (float)threadIdx.x / 16.0f, &s, &c);
        tw[threadIdx.x] = make_float2(c, s);
    }
    __syncthreads();

    const int idx = blockIdx.x * blockDim.x + threadIdx.x;
    if (idx >= Bb * Tt * Mm) return;
    const int m = idx % Mm;
    const int t = (idx / Mm) % Tt;
    const int b = idx / (Mm * Tt);

    // pre-shift gather: x[d'] = Ax[b, (d'+8)&15, t, m]
    float2 x[16];
    #pragma unroll
    for (int dp = 0; dp < 16; ++dp) {
        const int ds = (dp + 8) & 15;
        x[dp] = Ax[((size_t)(b * Dd + ds) * Tt + t) * Mm + m];
    }
    // out[dd] = sum_{d'} x[d'] * w^{((dd+8)&15) * d'},  w = e^{-2pi i/16}  (post-shift folded)
    #pragma unroll
    for (int dd = 0; dd < 16; ++dd) {
        const int k = (dd + 8) & 15;
        float re = 0.0f, im = 0.0f;
        #pragma unroll
        for (int dp = 0; dp < 16; ++dp) {
            const float2 w = tw[(k * dp) & 15];
            re += x[dp].x * w.x - x[dp].y * w.y;
            im += x[dp].x * w.y + x[dp].y * w.x;
        }
        out[((size_t)(b * Mm + m) * Dd + dd) * Tt + t] = make_float2(re, im);
    }
}

// ---------------------------------------------------------------------------
extern "C" void kernel_launch(void* const* d_in, const int* in_sizes, int n_in,
                              void* d_out, int out_size, void* d_ws, size_t ws_size,
                              hipStream_t stream) {
    const float2* o1   = (const float2*)d_in[0];  // complex64 [B,D,N,M]
    const float2* c1   = (const float2*)d_in[1];  // complex64 [B,M,D,N]
    const float*  m1   = (const float*)d_in[2];   // f32 [B,D,T]
    const float*  f1   = (const float*)d_in[3];   // f32 [B,D,N,1]
    const float2* fmt  = (const float2*)d_in[4];  // complex64 [T,N]
    const float*  tl   = (const float*)d_in[5];   // f32 [T]
    const int*    updn = (const int*)d_in[6];     // scalar

    float2* Ax  = (float2*)d_ws;                  // [B,D,T,M] complex, ~151 MB
    float2* out = (float2*)d_out;                 // [B,M,D,T] complex

    dim3 grid(Mm / TILE, Tt / TILE, Bb * Dd);     // (6, 6, 128)
    fmt_gemm_wmma_kernel<<<grid, 256, 0, stream>>>(o1, c1, m1, f1, fmt, tl, updn, Ax);

    const int total = Bb * Tt * Mm;
    dfft16_shift_kernel<<<(total + 255) / 256, 256, 0, stream>>>(Ax, out);
}